// GraphAttention_34754875359438
// MI455X (gfx1250) — compile-verified
//
#include <hip/hip_runtime.h>

typedef __attribute__((ext_vector_type(2))) float v2f;
typedef __attribute__((ext_vector_type(8))) float v8f;

namespace {
constexpr int kS = 64;     // neighbors per group
constexpr int kD = 128;    // feature dim
constexpr int kLdsStride = 144;  // 128 + 16 pad -> phase-3 lane halves hit disjoint bank sets
constexpr float kAlpha = 0.2f;   // LeakyReLU negative slope
}

__global__ __launch_bounds__(256) void gat_pool_kernel(
    const float* __restrict__ center,  // [N, 64, 128]
    const float* __restrict__ nbr,     // [N, 64, 128]
    const float* __restrict__ a,       // [256, 1]
    float* __restrict__ out)           // [N, 128]
{
    __shared__ float nbr_t[kS][kLdsStride];  // 36,864 B
    __shared__ float e_s[kS];
    __shared__ float att[kS];

    const int tid  = threadIdx.x;
    const int lane = tid & 31;
    const int wave = tid >> 5;
    const long long n = blockIdx.x;

    // Per-lane slice of the attention vector 'a' (L2-resident, 1 KB total).
    const float4 a1 = *(const float4*)(a +       lane * 4);
    const float4 a2 = *(const float4*)(a + kD +  lane * 4);

    const float* gc = center + n * (long long)(kS * kD);
    const float* gn = nbr    + n * (long long)(kS * kD);

    // ---------- Phase 1: stream both tensors once; scores + LDS stash ----------
    // Wave w handles rows s = w + 8k. Each row load = 32 lanes x float4 = 512 B coalesced.
    #pragma unroll
    for (int k = 0; k < 8; ++k) {
        const int s = wave + 8 * k;
        const float4 cv = *(const float4*)(gc + s * kD + lane * 4);
        const float4 nv = *(const float4*)(gn + s * kD + lane * 4);
        *(float4*)(&nbr_t[s][lane * 4]) = nv;  // keep nbr on-chip for phase 3
        float r = cv.x * a1.x + cv.y * a1.y + cv.z * a1.z + cv.w * a1.w
                + nv.x * a2.x + nv.y * a2.y + nv.z * a2.z + nv.w * a2.w;
        #pragma unroll
        for (int off = 16; off > 0; off >>= 1) r += __shfl_xor(r, off, 32);
        if (lane == 0) e_s[s] = r;
    }
    __syncthreads();

    // ---------- Phase 2: leaky-relu + softmax over 64 neighbors (wave 0) ----------
    if (wave == 0) {
        float e0 = e_s[lane];
        float e1 = e_s[lane + 32];
        e0 = e0 > 0.0f ? e0 : kAlpha * e0;
        e1 = e1 > 0.0f ? e1 : kAlpha * e1;
        float m = fmaxf(e0, e1);
        #pragma unroll
        for (int off = 16; off > 0; off >>= 1) m = fmaxf(m, __shfl_xor(m, off, 32));
        const float x0 = __expf(e0 - m);
        const float x1 = __expf(e1 - m);
        float ssum = x0 + x1;
        #pragma unroll
        for (int off = 16; off > 0; off >>= 1) ssum += __shfl_xor(ssum, off, 32);
        const float inv = 1.0f / ssum;
        att[lane]      = x0 * inv;
        att[lane + 32] = x1 * inv;
    }
    __syncthreads();

    // ---------- Phase 3: out = att[1x64] @ nbr[64x128] via V_WMMA_F32_16X16X4_F32 ----
    // A[m,k] = att[4kc+k] for all m (broadcast rows) => every row of D is the result.
    // B[k,col] = nbr_t[4kc+k][16*wave+col]. Wave w owns feature chunk [16w, 16w+16).
    // ISA layout (f32, 16x16x4): lanes 0-15 carry K={0,1}, lanes 16-31 carry K={2,3}
    // for both A (v0=K even, v1=K odd) and B.
    const int col  = lane & 15;
    const int koff = (lane >> 4) << 1;  // 0 for lanes 0-15, 2 for lanes 16-31
    const int fbase = 16 * wave;
    v8f acc = {};
    #pragma unroll
    for (int kc = 0; kc < 16; ++kc) {
        const int k0 = 4 * kc + koff;
        v2f av, bv;
        av.x = att[k0];
        av.y = att[k0 + 1];
        bv.x = nbr_t[k0][fbase + col];
        bv.y = nbr_t[k0 + 1][fbase + col];
        acc = __builtin_amdgcn_wmma_f32_16x16x4_f32(
            /*neg_a=*/false, av, /*neg_b=*/false, bv,
            /*c_mod=*/(short)0, acc, /*reuse_a=*/false, /*reuse_b=*/false);
    }
    // All 16 rows of D are identical; lanes 0-15, VGPR0 holds out[fbase + lane].
    if (lane < 16) out[n * kD + fbase + lane] = acc[0];
}

extern "C" void kernel_launch(void* const* d_in, const int* in_sizes, int n_in,
                              void* d_out, int out_size, void* d_ws, size_t ws_size,
                              hipStream_t stream) {
    (void)n_in; (void)out_size; (void)d_ws; (void)ws_size;
    const float* center = (const float*)d_in[0];
    const float* nbr    = (const float*)d_in[1];
    const float* a      = (const float*)d_in[2];
    float* out          = (float*)d_out;
    const int N = in_sizes[0] / (kS * kD);  // 8192 for the reference shapes
    if (N <= 0) return;
    gat_pool_kernel<<<dim3((unsigned)N), dim3(256), 0, stream>>>(center, nbr, a, out);
}